// Sim_Grasp_Net_19705309954200
// MI455X (gfx1250) — compile-verified
//
#include <hip/hip_runtime.h>

typedef __attribute__((ext_vector_type(2))) float v2f;
typedef __attribute__((ext_vector_type(8))) float v8f;

// ---- problem constants (from reference) ----
constexpr int Bb   = 2;
constexpr int Nn   = 20480;
constexpr int Cc   = 256;
constexpr int NDd  = 16384;
constexpr int NSs  = 512;
constexpr int Mm   = 2048;
constexpr int NVv  = 800;
constexpr int GS84 = 84;           // 12 angles * 7 depths

// output layout (flat, in return order)
constexpr size_t OFF_XYZ  = 0;                      // [2,2048,3]    = 12288
constexpr size_t OFF_FEAT = 12288;                  // [2,2048,256]  = 1048576
constexpr size_t OFF_AFF  = 1060864;                // [2,16384]     = 32768
constexpr size_t OFF_VS   = 1093632;                // [2,512,800]   = 819200
constexpr size_t OFF_GS   = 1912832;                // [2,512,800,84]= 68812800
constexpr size_t ZERO_F4  = (819200ull + 68812800ull) / 4;  // float4 count to zero

// =====================================================================
// Branchless selection helpers -> v_cndmask chains, no exec-mask branches
// =====================================================================
__device__ __forceinline__ bool better2(float d1, int i1, float d2, int i2) {
    return (d1 < d2) || (d1 == d2 && i1 < i2);     // smaller dist, tie -> lower idx
}
__device__ __forceinline__ bool betterMax(float d1, int i1, float d2, int i2) {
    return (d1 > d2) || (d1 == d2 && i1 < i2);     // larger val, tie -> lower idx
}
__device__ __forceinline__ void top2_insert(float d, int i,
                                            float& r1d, int& r1i, float& r2d, int& r2i) {
    const bool c1 = better2(d, i, r1d, r1i);
    const bool c2 = better2(d, i, r2d, r2i);
    const float n2d = c1 ? r1d : (c2 ? d : r2d);
    const int   n2i = c1 ? r1i : (c2 ? i : r2i);
    r1d = c1 ? d : r1d;
    r1i = c1 ? i : r1i;
    r2d = n2d;
    r2i = n2i;
}
__device__ __forceinline__ void top2_merge(float o1d, int o1i, float o2d, int o2i,
                                           float& r1d, int& r1i, float& r2d, int& r2i) {
    const bool c = better2(o1d, o1i, r1d, r1i);
    const float m1d = c ? o1d : r1d;  const int m1i = c ? o1i : r1i;
    const float ad  = c ? r1d : o1d;  const int ai  = c ? r1i : o1i;
    const float bd2 = c ? o2d : r2d;  const int bi2 = c ? o2i : r2i;
    const bool c2 = better2(ad, ai, bd2, bi2);
    r1d = m1d; r1i = m1i;
    r2d = c2 ? ad : bd2;
    r2i = c2 ? ai : bi2;
}
__device__ __forceinline__ void argmax_take(float od, int oi, float& bd, int& bi) {
    const bool c = betterMax(od, oi, bd, bi);
    bd = c ? od : bd;
    bi = c ? oi : bi;
}

// =====================================================================
// Zero-fill view_scores + grasp_scores region (283MB -> bandwidth floor)
// =====================================================================
__global__ void zero_kernel(float4* __restrict__ p, size_t n4) {
    size_t i = (size_t)blockIdx.x * blockDim.x + threadIdx.x;
    size_t stride = (size_t)gridDim.x * blockDim.x;
    float4 z; z.x = 0.f; z.y = 0.f; z.z = 0.f; z.w = 0.f;
    for (; i < n4; i += stride) p[i] = z;
}

// =====================================================================
// Masked furthest-point sampling: 1 block per batch, all state in VGPRs
// =====================================================================
constexpr int FPS_T = 1024;
constexpr int FPS_P = Nn / FPS_T;   // 20 points per thread

__global__ __launch_bounds__(FPS_T)
void fps_kernel(const float* __restrict__ xyz, const float* __restrict__ gsc,
                int* __restrict__ fps_idx) {
    const int b   = blockIdx.x;
    const int tid = threadIdx.x;
    const float* X = xyz + (size_t)b * Nn * 3;
    const float* G = gsc + (size_t)b * Nn;

    float px[FPS_P], py[FPS_P], pz[FPS_P], dist[FPS_P];
    unsigned mbits = 0u;
#pragma unroll
    for (int j = 0; j < FPS_P; ++j) {
        int i = tid + j * FPS_T;
        px[j] = X[i * 3 + 0]; py[j] = X[i * 3 + 1]; pz[j] = X[i * 3 + 2];
        bool v = G[i] > 0.001f;
        mbits |= (v ? 1u : 0u) << j;
        dist[j] = v ? 1e10f : -1.0f;
    }

    __shared__ float s_d[32];
    __shared__ int   s_i[32];
    __shared__ float s_bx, s_by, s_bz;
    __shared__ int   s_sel;

    const int wave = tid >> 5, lane = tid & 31;

    // start = argmax(mask) == first valid index (branchless min-index reduce)
    int fmin = 0x7fffffff;
#pragma unroll
    for (int j = 0; j < FPS_P; ++j) {
        int i = tid + j * FPS_T;
        bool v = (mbits >> j) & 1u;
        int cand = v ? i : 0x7fffffff;
        fmin = (cand < fmin) ? cand : fmin;
    }
    for (int o = 16; o > 0; o >>= 1) {
        int oi = __shfl_down(fmin, o, 32);
        fmin = (oi < fmin) ? oi : fmin;
    }
    if (lane == 0) s_i[wave] = fmin;
    __syncthreads();
    if (tid < 32) {
        int v = s_i[tid];
        for (int o = 16; o > 0; o >>= 1) {
            int oi = __shfl_down(v, o, 32);
            v = (oi < v) ? oi : v;
        }
        if (tid == 0) {
            int sel = (v == 0x7fffffff) ? 0 : v;
            s_sel = sel;
            s_bx = X[sel * 3 + 0]; s_by = X[sel * 3 + 1]; s_bz = X[sel * 3 + 2];
        }
    }
    __syncthreads();

    for (int m = 0; m < Mm; ++m) {
        const float lx = s_bx, ly = s_by, lz = s_bz;
        if (tid == 0) fps_idx[b * Mm + m] = s_sel;

        float bd = -2.0f; int bi = 0x7fffffff;
#pragma unroll
        for (int j = 0; j < FPS_P; ++j) {
            float dx = px[j] - lx, dy = py[j] - ly, dz = pz[j] - lz;
            float d = dx * dx + dy * dy + dz * dz;
            float nd = fminf(dist[j], d);
            dist[j] = ((mbits >> j) & 1u) ? nd : -1.0f;
            argmax_take(dist[j], tid + j * FPS_T, bd, bi);
        }
        for (int o = 16; o > 0; o >>= 1) {
            float od = __shfl_down(bd, o, 32);
            int   oi = __shfl_down(bi, o, 32);
            argmax_take(od, oi, bd, bi);
        }
        if (lane == 0) { s_d[wave] = bd; s_i[wave] = bi; }
        __syncthreads();
        if (tid < 32) {
            bd = s_d[tid]; bi = s_i[tid];
            for (int o = 16; o > 0; o >>= 1) {
                float od = __shfl_down(bd, o, 32);
                int   oi = __shfl_down(bi, o, 32);
                argmax_take(od, oi, bd, bi);
            }
            if (tid == 0) {
                s_sel = bi;
                s_bx = X[bi * 3 + 0]; s_by = X[bi * 3 + 1]; s_bz = X[bi * 3 + 2];
            }
        }
        __syncthreads();
    }
}

// =====================================================================
// Gather xyz + features at the FPS indices (block per (b,m), 256 = C)
// =====================================================================
__global__ void gather_kernel(const float* __restrict__ xyz, const float* __restrict__ feat,
                              const int* __restrict__ fps_idx,
                              float* __restrict__ o_xyz, float* __restrict__ o_feat) {
    const int bm = blockIdx.x;            // 0 .. B*M-1
    const int b  = bm / Mm;
    const int idx = fps_idx[bm];
    const int t  = threadIdx.x;
    const float* fsrc = feat + ((size_t)b * Nn + idx) * Cc;
    o_feat[(size_t)bm * Cc + t] = fsrc[t];
    if (t < 3) o_xyz[bm * 3 + t] = xyz[((size_t)b * Nn + idx) * 3 + t];
}

// =====================================================================
// Stage 2: d2 = |a|^2 + |s|^2 - 2 a.s  via V_WMMA_F32_16X16X4_F32 (K=3 pad 4),
// then per-row top-2 over 512 sparse points -> affordance = mean of 2-NN scores.
// Block = 256 threads (8 waves), wave owns one 16-row tile; grid = B * 128.
// =====================================================================
__global__ __launch_bounds__(256)
void stage2_kernel(const float* __restrict__ dense, const float* __restrict__ sparse,
                   const float* __restrict__ scores, float* __restrict__ afford) {
    __shared__ float s_sp[NSs * 4];   // x,y,z,|s|^2
    __shared__ float s_val[NSs];      // 1.0 valid / 0.0 invalid
    __shared__ float s_sc[NSs];

    const int b   = blockIdx.x >> 7;          // / 128
    const int blk = blockIdx.x & 127;
    const int tid = threadIdx.x;

    for (int m = tid; m < NSs; m += 256) {
        float x = sparse[((size_t)b * NSs + m) * 3 + 0];
        float y = sparse[((size_t)b * NSs + m) * 3 + 1];
        float z = sparse[((size_t)b * NSs + m) * 3 + 2];
        s_sp[m * 4 + 0] = x; s_sp[m * 4 + 1] = y; s_sp[m * 4 + 2] = z;
        s_sp[m * 4 + 3] = x * x + y * y + z * z;
        s_val[m] = ((x != 0.f) || (y != 0.f) || (z != 0.f)) ? 1.f : 0.f;
        s_sc[m]  = scores[(size_t)b * NSs + m];
    }
    __syncthreads();

    const int wave = tid >> 5, lane = tid & 31;
    const int half = lane >> 4, lm = lane & 15;
    const int rowBase = (blk * 8 + wave) * 16;          // dense row tile base
    const int r = rowBase + lm;

    const float* dp = dense + ((size_t)b * NDd + r) * 3;
    const float fx = dp[0], fy = dp[1], fz = dp[2];
    v2f A; A.x = half ? fz : fx; A.y = half ? 0.f : fy;  // A 16x4 layout, K=3 padded
    const float nrm = fx * fx + fy * fy + fz * fz;

    float na[8];
#pragma unroll
    for (int v = 0; v < 8; ++v) na[v] = __shfl(nrm, v + 8 * half, 32);

    float r1d[8], r2d[8]; int r1i[8], r2i[8];
#pragma unroll
    for (int v = 0; v < 8; ++v) { r1d[v] = 3e38f; r2d[v] = 3e38f; r1i[v] = 0x7fffffff; r2i[v] = 0x7fffffff; }

    for (int mt = 0; mt < NSs / 16; ++mt) {
        const int col = mt * 16 + lm;
        const float sx = s_sp[col * 4 + 0], sy = s_sp[col * 4 + 1], sz = s_sp[col * 4 + 2];
        const float nb = s_sp[col * 4 + 3];
        const float vv = s_val[col];
        v2f Bm; Bm.x = half ? sz : sx; Bm.y = half ? 0.f : sy;

        v8f acc = {};
        acc = __builtin_amdgcn_wmma_f32_16x16x4_f32(
            false, A, false, Bm, (short)0, acc, false, false);

#pragma unroll
        for (int v = 0; v < 8; ++v) {
            float d2 = na[v] + nb - 2.0f * acc[v];
            d2 = (vv > 0.f) ? d2 : 1e10f;
            top2_insert(d2, col, r1d[v], r1i[v], r2d[v], r2i[v]);
        }
    }

#pragma unroll
    for (int v = 0; v < 8; ++v) {
#pragma unroll
        for (int m = 1; m < 16; m <<= 1) {
            float o1d = __shfl_xor(r1d[v], m, 32); int o1i = __shfl_xor(r1i[v], m, 32);
            float o2d = __shfl_xor(r2d[v], m, 32); int o2i = __shfl_xor(r2i[v], m, 32);
            top2_merge(o1d, o1i, o2d, o2i, r1d[v], r1i[v], r2d[v], r2i[v]);
        }
        if (lm == v) {
            int row = rowBase + v + 8 * half;   // C/D layout: VGPR v, halves hold M=v / M=v+8
            afford[(size_t)b * NDd + row] = 0.5f * (s_sc[r1i[v]] + s_sc[r2i[v]]);
        }
    }
}

// =====================================================================
// Stage 3: argmax_v approach . view over 800 Fibonacci-sphere views,
// again V_WMMA_F32_16X16X4_F32. 3072 rows = 192 tiles; grid = 24 x 256.
// =====================================================================
__global__ __launch_bounds__(256)
void stage3_kernel(const float* __restrict__ adir, int* __restrict__ view_inds) {
    __shared__ float s_v[NVv * 4];
    const int tid = threadIdx.x;
    for (int i = tid; i < NVv; i += 256) {
        const double phi = 0.61803398874989484820458683436564;  // (sqrt(5)-1)/2
        double zi = (2.0 * i + 1.0) / (double)NVv - 1.0;
        double rr = sqrt(fmax(1.0 - zi * zi, 0.0));
        double th = 6.283185307179586476925286766559 * (double)i * phi;
        s_v[i * 4 + 0] = (float)(rr * cos(th));
        s_v[i * 4 + 1] = (float)(rr * sin(th));
        s_v[i * 4 + 2] = (float)zi;
        s_v[i * 4 + 3] = 0.f;
    }
    __syncthreads();

    const int wave = tid >> 5, lane = tid & 31;
    const int half = lane >> 4, lm = lane & 15;
    const int rowBase = (blockIdx.x * 8 + wave) * 16;    // flat row in [0, 3072)
    const int r = rowBase + lm;

    const float* ap = adir + (size_t)r * 3;
    v2f A; A.x = half ? ap[2] : ap[0]; A.y = half ? 0.f : ap[1];

    float bd[8]; int bi[8];
#pragma unroll
    for (int v = 0; v < 8; ++v) { bd[v] = -3e38f; bi[v] = 0x7fffffff; }

    for (int vt = 0; vt < NVv / 16; ++vt) {
        const int col = vt * 16 + lm;
        const float sx = s_v[col * 4 + 0], sy = s_v[col * 4 + 1], sz = s_v[col * 4 + 2];
        v2f Bm; Bm.x = half ? sz : sx; Bm.y = half ? 0.f : sy;

        v8f acc = {};
        acc = __builtin_amdgcn_wmma_f32_16x16x4_f32(
            false, A, false, Bm, (short)0, acc, false, false);

#pragma unroll
        for (int v = 0; v < 8; ++v) {
            argmax_take(acc[v], col, bd[v], bi[v]);
        }
    }

#pragma unroll
    for (int v = 0; v < 8; ++v) {
#pragma unroll
        for (int m = 1; m < 16; m <<= 1) {
            float od = __shfl_xor(bd[v], m, 32);
            int   oi = __shfl_xor(bi[v], m, 32);
            argmax_take(od, oi, bd[v], bi[v]);
        }
        if (lm == v) view_inds[rowBase + v + 8 * half] = bi[v];
    }
}

// =====================================================================
// Scatter view/grasp scores at the argmax view slots (after zero-fill)
// =====================================================================
__global__ void scatter_kernel(const int* __restrict__ view_inds, const float* __restrict__ sparse,
                               const float* __restrict__ nvs, const float* __restrict__ ngs,
                               float* __restrict__ vs_out, float* __restrict__ gs_out) {
    const int t = blockIdx.x * blockDim.x + threadIdx.x;
    const int R = Bb * NSs * 3;                // 3072
    if (t >= R * (GS84 + 1)) return;
    const int r = t / (GS84 + 1);
    const int j = t - r * (GS84 + 1);
    const int b = r / (NSs * 3);
    const int rem = r - b * NSs * 3;
    const int ns = rem / 3;

    const float* sp = sparse + ((size_t)b * NSs + ns) * 3;
    const float vm = ((sp[0] != 0.f) || (sp[1] != 0.f) || (sp[2] != 0.f)) ? 1.f : 0.f;
    const int v = view_inds[r];

    if (j == GS84) {
        vs_out[((size_t)b * NSs + ns) * NVv + v] = nvs[r] * vm;
    } else {
        gs_out[(((size_t)b * NSs + ns) * NVv + v) * GS84 + j] = ngs[(size_t)r * GS84 + j] * vm;
    }
}

// =====================================================================
extern "C" void kernel_launch(void* const* d_in, const int* in_sizes, int n_in,
                              void* d_out, int out_size, void* d_ws, size_t ws_size,
                              hipStream_t stream) {
    (void)in_sizes; (void)n_in; (void)out_size; (void)ws_size;
    const float* seed_xyz  = (const float*)d_in[0];
    const float* seed_feat = (const float*)d_in[1];
    const float* gsc       = (const float*)d_in[2];
    const float* dense     = (const float*)d_in[3];
    const float* sparse    = (const float*)d_in[4];
    const float* nscores   = (const float*)d_in[5];
    const float* adir      = (const float*)d_in[6];
    const float* nvs       = (const float*)d_in[7];
    const float* ngs       = (const float*)d_in[8];

    float* out    = (float*)d_out;
    float* o_xyz  = out + OFF_XYZ;
    float* o_feat = out + OFF_FEAT;
    float* o_aff  = out + OFF_AFF;
    float* o_vs   = out + OFF_VS;
    float* o_gs   = out + OFF_GS;

    int* fps_idx   = (int*)d_ws;                 // B*M   = 4096 ints
    int* view_inds = fps_idx + Bb * Mm;          // B*NS*3= 3072 ints

    zero_kernel<<<2048, 256, 0, stream>>>((float4*)o_vs, ZERO_F4);
    fps_kernel<<<Bb, FPS_T, 0, stream>>>(seed_xyz, gsc, fps_idx);
    gather_kernel<<<Bb * Mm, Cc, 0, stream>>>(seed_xyz, seed_feat, fps_idx, o_xyz, o_feat);
    stage2_kernel<<<Bb * (NDd / (16 * 8)), 256, 0, stream>>>(dense, sparse, nscores, o_aff);
    stage3_kernel<<<(Bb * NSs * 3) / (16 * 8), 256, 0, stream>>>(adir, view_inds);
    scatter_kernel<<<(Bb * NSs * 3 * (GS84 + 1) + 255) / 256, 256, 0, stream>>>(
        view_inds, sparse, nvs, ngs, o_vs, o_gs);
}